// UpperGraphEncoder_16063177687557
// MI455X (gfx1250) — compile-verified
//
#include <hip/hip_runtime.h>
#include <hip/hip_bf16.h>

typedef __attribute__((ext_vector_type(2))) float v2f;
typedef __attribute__((ext_vector_type(8))) float v8f;

#define F_DIM 64
#define H_DIM 128

// ---------------------------------------------------------------------------
// WMMA tiling (shared by both GEMMs), V_WMMA_F32_16X16X4_F32, wave32:
//   Block = 256 threads = 8 waves, covers 64 rows x 128 cols.
//   Wave w: row tile = w>>1 (16 rows), col group = w&1 (64 cols = 4 tiles).
// Fragment layouts (ISA 7.12.2, 32-bit):
//   A 16x4 : lane(0-15)=M, kh=lane>>4 ; v0 = A[M][k0+2*kh], v1 = A[M][k0+2*kh+1]
//   B 4x16 : lane(0-15)=N (mirror)    ; v0 = B[k0+2*kh][N], v1 = B[k0+2*kh+1][N]
//   C/D    : VGPR i -> M = i + 8*kh, N = lane&15
// ---------------------------------------------------------------------------

template <int K>
__device__ __forceinline__ void wmma_mainloop(
    const float* __restrict__ Ar,      // A row pointer for this lane (stride K)
    const float* __restrict__ W,       // (K x 128) weights
    int cgrp, int nsub, int kh, v8f acc[4])
{
  for (int kk = 0; kk < K; kk += 4) {
    v2f a;
    a.x = Ar[kk + 2 * kh + 0];
    a.y = Ar[kk + 2 * kh + 1];
#pragma unroll
    for (int c = 0; c < 4; ++c) {
      int n = cgrp * 64 + c * 16 + nsub;
      v2f b;
      b.x = W[(size_t)(kk + 2 * kh + 0) * H_DIM + n];
      b.y = W[(size_t)(kk + 2 * kh + 1) * H_DIM + n];
      acc[c] = __builtin_amdgcn_wmma_f32_16x16x4_f32(
          false, a, false, b, (short)0, acc[c], false, false);
    }
  }
}

// Epilogue: h = relu(acc + bias)  (RESIDUAL=false)  or  h += relu(acc + bias).
// GUARD=false is the block-uniform fast path: straight-line stores, no exec
// masking (the per-element r<N guard otherwise costs a saveexec per store).
template <bool RESIDUAL, bool GUARD>
__device__ __forceinline__ void wmma_epilogue(
    const v8f acc[4], const float* __restrict__ bias, float* __restrict__ h,
    int rbase, int cgrp, int nsub, int N)
{
#pragma unroll
  for (int c = 0; c < 4; ++c) {
    int n = cgrp * 64 + c * 16 + nsub;
    float bb = bias[n];
#pragma unroll
    for (int i = 0; i < 8; ++i) {
      int r = rbase + i;
      if (!GUARD || r < N) {
        size_t idx = (size_t)r * H_DIM + n;
        float v = acc[c][i] + bb;
        v = v > 0.f ? v : 0.f;
        h[idx] = RESIDUAL ? (h[idx] + v) : v;
      }
    }
  }
}

// ---------------------------------------------------------------------------
// GEMM 1: h = relu(X @ W_in + b_in)      X:(N,64)  W:(64,128)  h:(N,128)
// ---------------------------------------------------------------------------
__global__ __launch_bounds__(256) void gemm_in_kernel(
    const float* __restrict__ X, const float* __restrict__ W,
    const float* __restrict__ bias, float* __restrict__ h, int N)
{
  const int lane  = threadIdx.x & 31;
  const int wave  = threadIdx.x >> 5;
  const int rtile = wave >> 1;
  const int cgrp  = wave & 1;
  const int nsub  = lane & 15;
  const int kh    = lane >> 4;

  const bool full = (blockIdx.x + 1) * 64 <= (unsigned)N;   // block-uniform
  int row  = blockIdx.x * 64 + rtile * 16 + nsub;
  int arow = full ? row : (row < N ? row : N - 1);

  v8f acc[4] = {};
  wmma_mainloop<F_DIM>(X + (size_t)arow * F_DIM, W, cgrp, nsub, kh, acc);

  const int rbase = blockIdx.x * 64 + rtile * 16 + 8 * kh;
  if (full) wmma_epilogue<false, false>(acc, bias, h, rbase, cgrp, nsub, N);
  else      wmma_epilogue<false, true >(acc, bias, h, rbase, cgrp, nsub, N);
}

// ---------------------------------------------------------------------------
// GEMM 2/3: h = h + relu(msg @ W + b)    msg:(N,128)  W:(128,128)
// In-place on h is safe: each (r,n) is read+written by exactly one thread.
// ---------------------------------------------------------------------------
__global__ __launch_bounds__(256) void gemm_layer_kernel(
    const float* __restrict__ msg, const float* __restrict__ W,
    const float* __restrict__ bias, float* __restrict__ h, int N)
{
  const int lane  = threadIdx.x & 31;
  const int wave  = threadIdx.x >> 5;
  const int rtile = wave >> 1;
  const int cgrp  = wave & 1;
  const int nsub  = lane & 15;
  const int kh    = lane >> 4;

  const bool full = (blockIdx.x + 1) * 64 <= (unsigned)N;
  int row  = blockIdx.x * 64 + rtile * 16 + nsub;
  int arow = full ? row : (row < N ? row : N - 1);

  v8f acc[4] = {};
  wmma_mainloop<H_DIM>(msg + (size_t)arow * H_DIM, W, cgrp, nsub, kh, acc);

  const int rbase = blockIdx.x * 64 + rtile * 16 + 8 * kh;
  if (full) wmma_epilogue<true, false>(acc, bias, h, rbase, cgrp, nsub, N);
  else      wmma_epilogue<true, true >(acc, bias, h, rbase, cgrp, nsub, N);
}

// ---------------------------------------------------------------------------
// Scatter-add: msg[dst[e]] += h[src[e]].  One wave32 per edge; lanes cover the
// 128 features in 4 coalesced 128B sweeps. h and msg are L2-resident (51 MB
// each vs 192 MB L2), so gathers + float atomics stay on-chip.
// ---------------------------------------------------------------------------
__global__ __launch_bounds__(256) void scatter_kernel(
    const float* __restrict__ h, float* __restrict__ msg,
    const int* __restrict__ src, const int* __restrict__ dst, int nE)
{
  int t = blockIdx.x * blockDim.x + threadIdx.x;
  int e = t >> 5;
  int lane = t & 31;
  if (e >= nE) return;
  int s = src[e];
  int d = dst[e];
  const float* hs = h + (size_t)s * H_DIM;
  float* md = msg + (size_t)d * H_DIM;
#pragma unroll
  for (int j = 0; j < 4; ++j) {
    int f = lane + 32 * j;
    atomicAdd(&md[f], hs[f]);
  }
}

// ---------------------------------------------------------------------------
// Column mean: out[c] = mean over N rows of h[:, c].
// ---------------------------------------------------------------------------
__global__ __launch_bounds__(128) void mean_kernel(
    const float* __restrict__ h, float* __restrict__ out, int N)
{
  int c = threadIdx.x;                 // 0..127
  float s = 0.f;
  for (int r = blockIdx.x; r < N; r += gridDim.x)
    s += h[(size_t)r * H_DIM + c];
  atomicAdd(&out[c], s * (1.0f / (float)N));
}

// ---------------------------------------------------------------------------
extern "C" void kernel_launch(void* const* d_in, const int* in_sizes, int n_in,
                              void* d_out, int out_size, void* d_ws, size_t ws_size,
                              hipStream_t stream)
{
  const float* X   = (const float*)d_in[0];
  const float* Win = (const float*)d_in[1];
  const float* bin = (const float*)d_in[2];
  const float* W1  = (const float*)d_in[3];
  const float* b1  = (const float*)d_in[4];
  const float* W2  = (const float*)d_in[5];
  const float* b2  = (const float*)d_in[6];
  const int*   ei  = (const int*)d_in[7];   // (2, nE): row 0 = src, row 1 = dst

  const int N  = in_sizes[0] / F_DIM;
  const int nE = in_sizes[7] / 2;

  float* h   = (float*)d_ws;                         // N*128 floats
  float* msg = h + (size_t)N * H_DIM;                // N*128 floats

  const int gRows = (N + 63) / 64;
  const int gScat = (int)(((size_t)nE * 32 + 255) / 256);

  gemm_in_kernel<<<gRows, 256, 0, stream>>>(X, Win, bin, h, N);

  const float* Ws[2] = {W1, W2};
  const float* bs[2] = {b1, b2};
  for (int l = 0; l < 2; ++l) {
    hipMemsetAsync(msg, 0, (size_t)N * H_DIM * sizeof(float), stream);
    scatter_kernel<<<gScat, 256, 0, stream>>>(h, msg, ei, ei + nE, nE);
    gemm_layer_kernel<<<gRows, 256, 0, stream>>>(msg, Ws[l], bs[l], h, N);
  }

  hipMemsetAsync(d_out, 0, H_DIM * sizeof(float), stream);
  mean_kernel<<<512, 128, 0, stream>>>(h, (float*)d_out, N);
}